// CriticalityLoss_28845000360534
// MI455X (gfx1250) — compile-verified
//
#include <hip/hip_runtime.h>
#include <math.h>

typedef __attribute__((ext_vector_type(2))) float v2f;
typedef __attribute__((ext_vector_type(8))) float v8f;

#define TPB 256
#define IPTA 8
#define SORT_IPT 8
#define SORT_TILE (TPB * SORT_IPT)   // 2048
#define CHUNK 4096                   // TPB * 16

struct Accum {
  double comp, mt, cons, ssum, cnt;
  unsigned maxbits;
  unsigned pad;
};

// Exact 32-lane f32 sum using V_WMMA_F32_16X16X4_F32 with B = ones.
// Stage 1: A[m][0]=x[m] (lanes 0-15), A[m][2]=x[m+16] (lanes 16-31), A[m][1]=A[m][3]=0
//   -> D[m][n] = x[m] + x[m+16] = y[m]; lane L holds y[0..7] or y[8..15] in its 8 D VGPRs.
// Stage 2: same trick on z = sum of D VGPRs -> every lane gets the full wave sum.
__device__ __forceinline__ float wave_reduce_sum_wmma(float x) {
  v2f a; a[0] = x; a[1] = 0.0f;
  v2f ones; ones[0] = 1.0f; ones[1] = 1.0f;
  v8f c = {};
  v8f d = __builtin_amdgcn_wmma_f32_16x16x4_f32(false, a, false, ones, (short)0, c,
                                                false, false);
  float z = d[0] + d[1] + d[2] + d[3] + d[4] + d[5] + d[6] + d[7];
  a[0] = z; a[1] = 0.0f;
  d = __builtin_amdgcn_wmma_f32_16x16x4_f32(false, a, false, ones, (short)0, c,
                                            false, false);
  return d[0];
}

__device__ __forceinline__ unsigned umaxu(unsigned a, unsigned b) { return a > b ? a : b; }

// ---------------- Phase A: streaming reductions + sort-key emission ----------------
__global__ void __launch_bounds__(TPB)
phaseA(const float* __restrict__ pred, const float* __restrict__ target,
       const int* __restrict__ mask, const float* __restrict__ rmav,
       unsigned* __restrict__ K0, float* __restrict__ V0,
       double* __restrict__ blockP, unsigned* __restrict__ blockMax, int n) {
  int base = blockIdx.x * (TPB * IPTA);
  float comp = 0.f, mt = 0.f, cons = 0.f, ssum = 0.f, cntf = 0.f;
  unsigned lmax = 0u;
  for (int k = 0; k < IPTA; ++k) {
    int r = base + k * TPB + threadIdx.x;
    if (r < n) {
      // prefetch next tile while this one streams (global_prefetch_b8)
      int rn = r + TPB;
      if (rn < n) {
        __builtin_prefetch(pred + (size_t)rn * 8, 0, 1);
        __builtin_prefetch(target + (size_t)rn * 8, 0, 1);
        __builtin_prefetch(rmav + (size_t)rn * 8, 0, 1);
      }
      bool m = mask[r] != 0;
      const float4* pp = (const float4*)(pred + (size_t)r * 8);
      const float4* tt = (const float4*)(target + (size_t)r * 8);
      const float4* rr = (const float4*)(rmav + (size_t)r * 8);
      float4 p0 = pp[0], p1 = pp[1];
      float4 t0 = tt[0], t1 = tt[1];
      float4 r0 = rr[0], r1 = rr[1];
      float d0 = p0.x - t0.x;
      float dy = p0.y - t0.y, dz = p0.z - t0.z, dw = p0.w - t0.w;
      float e0 = p1.x - t1.x, e1 = p1.y - t1.y, e2 = p1.z - t1.z, e3 = p1.w - t1.w;
      float aux = dy*dy + dz*dz + dw*dw + e0*e0 + e1*e1 + e2*e2 + e3*e3;
      float cy = p0.y - r0.y, cz = p0.z - r0.z, cw = p0.w - r0.w;
      float f0 = p1.x - r1.x, f1 = p1.y - r1.y, f2 = p1.z - r1.z, f3 = p1.w - r1.w;
      float ca = cy*cy + cz*cz + cw*cw + f0*f0 + f1*f1 + f2*f2 + f3*f3;
      if (m) {
        comp += d0 * d0; mt += aux; ssum += p0.x; cntf += 1.0f;
        unsigned su = __float_as_uint(p0.x);
        su = (su & 0x80000000u) ? ~su : (su | 0x80000000u);
        lmax = umaxu(lmax, su);
      } else {
        cons += ca;
      }
      float keyf = m ? t0.x : -1e30f;
      unsigned u = __float_as_uint(keyf);
      unsigned asc = (u & 0x80000000u) ? ~u : (u | 0x80000000u);
      K0[r] = ~asc;        // ascending uint == descending float
      V0[r] = p0.x;
    }
  }
  float wcomp = wave_reduce_sum_wmma(comp);
  float wmt   = wave_reduce_sum_wmma(mt);
  float wcons = wave_reduce_sum_wmma(cons);
  float wssum = wave_reduce_sum_wmma(ssum);
  float wcnt  = wave_reduce_sum_wmma(cntf);
  unsigned wm = lmax;
  for (int s = 16; s >= 1; s >>= 1) wm = umaxu(wm, (unsigned)__shfl_xor((int)wm, s));
  __shared__ float sc[8][5];
  __shared__ unsigned sm[8];
  int wave = threadIdx.x >> 5, lane = threadIdx.x & 31;
  if (lane == 0) {
    sc[wave][0] = wcomp; sc[wave][1] = wmt; sc[wave][2] = wcons;
    sc[wave][3] = wssum; sc[wave][4] = wcnt; sm[wave] = wm;
  }
  __syncthreads();
  if (threadIdx.x == 0) {
    double a0=0,a1=0,a2=0,a3=0,a4=0; unsigned mx=0;
    for (int w = 0; w < 8; ++w) {
      a0+=sc[w][0]; a1+=sc[w][1]; a2+=sc[w][2]; a3+=sc[w][3]; a4+=sc[w][4];
      mx = umaxu(mx, sm[w]);
    }
    double* bp = blockP + (size_t)blockIdx.x * 5;
    bp[0]=a0; bp[1]=a1; bp[2]=a2; bp[3]=a3; bp[4]=a4;
    blockMax[blockIdx.x] = mx;
  }
}

__global__ void __launch_bounds__(TPB)
reduceA(const double* __restrict__ blockP, const unsigned* __restrict__ blockMax,
        int nblocks, Accum* __restrict__ acc) {
  __shared__ double sd[TPB * 5];
  __shared__ unsigned su[TPB];
  double a[5] = {0,0,0,0,0}; unsigned mx = 0;
  for (int b = threadIdx.x; b < nblocks; b += TPB) {
    const double* bp = blockP + (size_t)b * 5;
    for (int q = 0; q < 5; ++q) a[q] += bp[q];
    mx = umaxu(mx, blockMax[b]);
  }
  for (int q = 0; q < 5; ++q) sd[q * TPB + threadIdx.x] = a[q];
  su[threadIdx.x] = mx;
  __syncthreads();
  for (int s = TPB / 2; s > 0; s >>= 1) {
    if (threadIdx.x < s) {
      for (int q = 0; q < 5; ++q) sd[q*TPB+threadIdx.x] += sd[q*TPB+threadIdx.x+s];
      su[threadIdx.x] = umaxu(su[threadIdx.x], su[threadIdx.x + s]);
    }
    __syncthreads();
  }
  if (threadIdx.x == 0) {
    acc->comp = sd[0]; acc->mt = sd[TPB]; acc->cons = sd[2*TPB];
    acc->ssum = sd[3*TPB]; acc->cnt = sd[4*TPB]; acc->maxbits = su[0];
  }
}

// ---------------- Stable LSD radix sort (4 x 8-bit) ----------------
__global__ void __launch_bounds__(TPB)
radixHist(const unsigned* __restrict__ keys, unsigned* __restrict__ hist,
          int n, int shift, int numTiles) {
  __shared__ unsigned h[256];
  h[threadIdx.x] = 0;
  __syncthreads();
  int base = blockIdx.x * SORT_TILE;
  for (int j = 0; j < SORT_IPT; ++j) {
    int i = base + j * TPB + threadIdx.x;
    if (i < n) atomicAdd(&h[(keys[i] >> shift) & 255u], 1u);
  }
  __syncthreads();
  hist[(size_t)threadIdx.x * numTiles + blockIdx.x] = h[threadIdx.x];
}

__global__ void __launch_bounds__(TPB)
radixScan(unsigned* __restrict__ hist, int numTiles) {
  __shared__ unsigned tot[256];
  int d = threadIdx.x;
  size_t off = (size_t)d * numTiles;
  unsigned run = 0;
  for (int t = 0; t < numTiles; ++t) { unsigned v = hist[off+t]; hist[off+t] = run; run += v; }
  tot[d] = run;
  __syncthreads();
  if (d == 0) { unsigned r = 0; for (int i = 0; i < 256; ++i) { unsigned v = tot[i]; tot[i] = r; r += v; } }
  __syncthreads();
  unsigned basev = tot[d];
  for (int t = 0; t < numTiles; ++t) hist[off + t] += basev;
}

// Stable scatter: rank = earlier-rounds count + earlier-waves count + ballot rank.
__global__ void __launch_bounds__(TPB)
radixScatter(const unsigned* __restrict__ keys, const float* __restrict__ vals,
             unsigned* __restrict__ okeys, float* __restrict__ ovals,
             const unsigned* __restrict__ hist, int n, int shift, int numTiles) {
  __shared__ unsigned short runhist[256];
  __shared__ unsigned short wcnt[8][256];
  int tid = threadIdx.x, wave = tid >> 5, lane = tid & 31;
  runhist[tid] = 0;
  unsigned* wz = (unsigned*)wcnt;
  for (int i = tid; i < 8 * 128; i += TPB) wz[i] = 0;
  __syncthreads();
  int base = blockIdx.x * SORT_TILE;
  unsigned lanes_below = (lane == 0) ? 0u : ((1u << lane) - 1u);
  for (int j = 0; j < SORT_IPT; ++j) {
    int i = base + j * TPB + tid;
    bool valid = i < n;
    unsigned k = valid ? keys[i] : 0u;
    float v = valid ? vals[i] : 0.f;
    unsigned d = (k >> shift) & 255u;
    unsigned mask = __builtin_amdgcn_ballot_w32(valid);
    for (int b = 0; b < 8; ++b) {
      unsigned bb = __builtin_amdgcn_ballot_w32(((d >> b) & 1u) != 0u);
      mask &= ((d >> b) & 1u) ? bb : ~bb;
    }
    unsigned rankInWave = __popc(mask & lanes_below);
    if (valid && rankInWave == 0) wcnt[wave][d] = (unsigned short)__popc(mask);
    __syncthreads();
    if (valid) {
      unsigned before = 0;
      for (int w = 0; w < wave; ++w) before += wcnt[w][d];
      unsigned pos = hist[(size_t)d * numTiles + blockIdx.x] +
                     (unsigned)runhist[d] + before + rankInWave;
      okeys[pos] = k; ovals[pos] = v;
    }
    __syncthreads();
    unsigned add = 0;
    for (int w = 0; w < 8; ++w) add += wcnt[w][tid];
    runhist[tid] = (unsigned short)(runhist[tid] + add);
    __syncthreads();
    for (int ii = tid; ii < 8 * 128; ii += TPB) wz[ii] = 0;
    __syncthreads();
  }
}

// ---------------- Suffix logsumexp over sorted labeled scores ----------------
__device__ __forceinline__ float decode_max(unsigned mb) {
  return __uint_as_float((mb & 0x80000000u) ? (mb ^ 0x80000000u) : ~mb);
}

__global__ void __launch_bounds__(TPB)
suffixE1(const float* __restrict__ svals, const Accum* __restrict__ acc,
         double* __restrict__ chunkSum, int n) {
  long cnt = (long)(acc->cnt + 0.5);
  double M = (double)decode_max(acc->maxbits);
  long base = (long)blockIdx.x * CHUNK;
  double s = 0.0;
  for (int j = 0; j < 16; ++j) {
    long i = base + j * TPB + threadIdx.x;
    if (i < cnt) s += exp((double)svals[i] - M);
  }
  __shared__ double sd[TPB];
  sd[threadIdx.x] = s; __syncthreads();
  for (int st = TPB / 2; st > 0; st >>= 1) {
    if (threadIdx.x < st) sd[threadIdx.x] += sd[threadIdx.x + st];
    __syncthreads();
  }
  if (threadIdx.x == 0) chunkSum[blockIdx.x] = sd[0];
}

__global__ void __launch_bounds__(TPB)
suffixE2(const double* __restrict__ cs, double* __restrict__ suf, int nch) {
  __shared__ double sh[2048];
  if (nch <= 2048) {
    for (int i = threadIdx.x; i < nch; i += TPB) sh[i] = cs[i];
    __syncthreads();
    if (threadIdx.x == 0) {
      double run = 0; for (int b = nch - 1; b >= 0; --b) { double v = sh[b]; sh[b] = run; run += v; }
    }
    __syncthreads();
    for (int i = threadIdx.x; i < nch; i += TPB) suf[i] = sh[i];
  } else if (threadIdx.x == 0) {
    double run = 0; for (int b = nch - 1; b >= 0; --b) { double v = cs[b]; suf[b] = run; run += v; }
  }
}

// Chunk staged into LDS via gfx1250 async global->LDS copies (ASYNCcnt),
// so per-instruction global traffic is fully coalesced (16B/lane, 512B/wave),
// then each thread reads its 16 contiguous floats from LDS.
// NOTE: reads up to CHUNK floats from svals even past n; svals (V0) is followed
// by the K1 workspace region, so the addresses stay inside d_ws (values unused).
__global__ void __launch_bounds__(TPB)
suffixE3(const float* __restrict__ svals, const Accum* __restrict__ acc,
         const double* __restrict__ chunkSuf, double* __restrict__ chunkLog, int n) {
  __shared__ float sbuf[CHUNK];
  long cnt = (long)(acc->cnt + 0.5);
  double M = (double)decode_max(acc->maxbits);
  long cbase = (long)blockIdx.x * CHUNK;

  for (int rr = 0; rr < CHUNK / (TPB * 4); ++rr) {   // 4 rounds of 16B per lane
    int g = rr * TPB + threadIdx.x;                  // 16-byte group index
    unsigned long long ga = (unsigned long long)(const void*)(svals + cbase + (long)g * 4);
    unsigned la = (unsigned)(size_t)(&sbuf[g * 4]);
    asm volatile("global_load_async_to_lds_b128 %0, %1, off"
                 :: "v"(la), "v"(ga) : "memory");
  }
  asm volatile("s_wait_asynccnt 0x0" ::: "memory");
  __syncthreads();

  long base = cbase + (long)threadIdx.x * 16;
  double e[16]; double tsum = 0.0;
  for (int j = 0; j < 16; ++j) {
    long i = base + j;
    double ev = (i < cnt) ? exp((double)sbuf[threadIdx.x * 16 + j] - M) : 0.0;
    e[j] = ev; tsum += ev;
  }
  __shared__ double sh[TPB];
  sh[threadIdx.x] = tsum; __syncthreads();
  for (int s = 1; s < TPB; s <<= 1) {          // inclusive suffix scan
    double v = (threadIdx.x + s < TPB) ? sh[threadIdx.x + s] : 0.0;
    __syncthreads();
    sh[threadIdx.x] += v;
    __syncthreads();
  }
  double run = ((threadIdx.x + 1 < TPB) ? sh[threadIdx.x + 1] : 0.0) + chunkSuf[blockIdx.x];
  double logacc = 0.0;
  for (int j = 15; j >= 0; --j) {
    run += e[j];
    if (base + j < cnt) logacc += log(run);
  }
  float wsum = wave_reduce_sum_wmma((float)logacc);
  __shared__ float wl[8];
  if ((threadIdx.x & 31) == 0) wl[threadIdx.x >> 5] = wsum;
  __syncthreads();
  if (threadIdx.x == 0) { double t = 0; for (int w = 0; w < 8; ++w) t += wl[w]; chunkLog[blockIdx.x] = t; }
}

__global__ void __launch_bounds__(TPB)
finalK(const Accum* __restrict__ acc, const double* __restrict__ chunkLog, int nch,
       float* __restrict__ out, int n) {
  __shared__ double sd[TPB];
  double s = 0.0;
  for (int i = threadIdx.x; i < nch; i += TPB) s += chunkLog[i];
  sd[threadIdx.x] = s; __syncthreads();
  for (int st = TPB / 2; st > 0; st >>= 1) {
    if (threadIdx.x < st) sd[threadIdx.x] += sd[threadIdx.x + st];
    __syncthreads();
  }
  if (threadIdx.x == 0) {
    double logT = sd[0];
    double cnt = acc->cnt;
    double ucnt = (double)n - cnt;
    double M = (double)decode_max(acc->maxbits);
    double comp = acc->comp / cnt;
    double mt = acc->mt / (cnt * 7.0);
    double cons = acc->cons / (ucnt * 7.0);
    double lpsum = acc->ssum - cnt * M - logT;
    double rank = -lpsum / cnt;
    double total = comp + 0.5 * mt + 0.3 * rank + 0.1 * cons;
    out[0] = (float)total; out[1] = (float)comp; out[2] = (float)mt;
    out[3] = (float)rank;  out[4] = (float)cons;
  }
}

extern "C" void kernel_launch(void* const* d_in, const int* in_sizes, int n_in,
                              void* d_out, int out_size, void* d_ws, size_t ws_size,
                              hipStream_t stream) {
  const float* pred   = (const float*)d_in[0];
  const float* target = (const float*)d_in[1];
  const int*   mask   = (const int*)d_in[2];
  const float* rmav   = (const float*)d_in[3];
  int n = in_sizes[0] / 8;
  int gridA     = (n + TPB * IPTA - 1) / (TPB * IPTA);
  int numTiles  = (n + SORT_TILE - 1) / SORT_TILE;
  int numChunks = (n + CHUNK - 1) / CHUNK;

  char* w = (char*)d_ws; size_t o = 0;
  unsigned* K0 = (unsigned*)(w + o); o += (size_t)n * 4;
  float*    V0 = (float*)(w + o);    o += (size_t)n * 4;
  unsigned* K1 = (unsigned*)(w + o); o += (size_t)n * 4;
  float*    V1 = (float*)(w + o);    o += (size_t)n * 4;
  unsigned* hist = (unsigned*)(w + o); o += (size_t)256 * numTiles * 4;
  o = (o + 7) & ~(size_t)7;
  double* blockP = (double*)(w + o); o += (size_t)gridA * 5 * 8;
  unsigned* blockMax = (unsigned*)(w + o); o += (size_t)gridA * 4;
  o = (o + 7) & ~(size_t)7;
  double* chunkSum = (double*)(w + o); o += (size_t)numChunks * 8;
  double* chunkSuf = (double*)(w + o); o += (size_t)numChunks * 8;
  double* chunkLog = (double*)(w + o); o += (size_t)numChunks * 8;
  Accum* acc = (Accum*)(w + o); o += sizeof(Accum);

  phaseA<<<gridA, TPB, 0, stream>>>(pred, target, mask, rmav, K0, V0, blockP, blockMax, n);
  reduceA<<<1, TPB, 0, stream>>>(blockP, blockMax, gridA, acc);

  unsigned* ks = K0; float* vs = V0; unsigned* kd = K1; float* vd = V1;
  for (int p = 0; p < 4; ++p) {
    radixHist<<<numTiles, TPB, 0, stream>>>(ks, hist, n, p * 8, numTiles);
    radixScan<<<1, TPB, 0, stream>>>(hist, numTiles);
    radixScatter<<<numTiles, TPB, 0, stream>>>(ks, vs, kd, vd, hist, n, p * 8, numTiles);
    unsigned* tk = ks; ks = kd; kd = tk;
    float* tv = vs; vs = vd; vd = tv;
  }
  // after 4 passes sorted data is back in K0/V0 (== ks/vs)
  suffixE1<<<numChunks, TPB, 0, stream>>>(vs, acc, chunkSum, n);
  suffixE2<<<1, TPB, 0, stream>>>(chunkSum, chunkSuf, numChunks);
  suffixE3<<<numChunks, TPB, 0, stream>>>(vs, acc, chunkSuf, chunkLog, n);
  finalK<<<1, TPB, 0, stream>>>(acc, chunkLog, numChunks, (float*)d_out, n);
}